// ExpertMLPs_20933670601018
// MI455X (gfx1250) — compile-verified
//
#include <hip/hip_runtime.h>
#include <math.h>

// Problem constants (from reference)
#define T_ 8192
#define H_ 1024
#define I_ 4096
#define E_ 8
#define K_ 2
#define C_ 2048   // ceil(T*K/E)

#define ASTRIDE 36    // A tile row stride in floats (32 + 4 pad); 144B, 16B aligned
#define BSTRIDE 132   // B tile row stride in floats (128 + 4 pad); 528B, 16B aligned

typedef __attribute__((ext_vector_type(2))) float v2f;
typedef __attribute__((ext_vector_type(8))) float v8f;
typedef int v4i __attribute__((vector_size(16)));   // matches builtin param type

typedef __attribute__((address_space(1))) v4i* gptr_v4i;   // global (__device__)
typedef __attribute__((address_space(3))) v4i* lptr_v4i;   // LDS (__shared__)

// Async global->LDS staging (gfx1250 ASYNCcnt path), guarded so a missing
// builtin falls back to the synchronous float4 path.
#if defined(__has_builtin)
#  if __has_builtin(__builtin_amdgcn_global_load_async_to_lds_b128) && \
      __has_builtin(__builtin_amdgcn_s_wait_asynccnt)
#    define MOE_ASYNC_LDS 1
#  endif
#endif
#ifndef MOE_ASYNC_LDS
#  define MOE_ASYNC_LDS 0
#endif

__device__ __forceinline__ void stage16(const float* __restrict__ g, float* l) {
#if MOE_ASYNC_LDS
    __builtin_amdgcn_global_load_async_to_lds_b128(
        (gptr_v4i)g, (lptr_v4i)l, 0, 0);
#else
    *(float4*)l = *(const float4*)g;
#endif
}

__device__ __forceinline__ void stage_fence() {
#if MOE_ASYNC_LDS
    __builtin_amdgcn_s_wait_asynccnt(0);
#endif
    __syncthreads();
}

// ---------------------------------------------------------------------------
// Kernel 0: zero output, init assignments to -1, zero scale slots
// ---------------------------------------------------------------------------
__global__ void moe_init_kernel(float* __restrict__ out, int* __restrict__ assign,
                                float* __restrict__ scale_slot) {
    size_t total = (size_t)T_ * H_;
    for (size_t i = (size_t)blockIdx.x * blockDim.x + threadIdx.x; i < total;
         i += (size_t)gridDim.x * blockDim.x)
        out[i] = 0.0f;
    int idx = blockIdx.x * blockDim.x + threadIdx.x;
    if (idx < E_ * C_) { assign[idx] = -1; scale_slot[idx] = 0.0f; }
}

// ---------------------------------------------------------------------------
// Kernel 1: routing. One wave32 per expert; token-order inclusive prefix sum
// via ballot so positions match the reference cumsum exactly.
// ---------------------------------------------------------------------------
__global__ void moe_route_kernel(const int* __restrict__ eidx,
                                 int* __restrict__ assign,
                                 int* __restrict__ tok_slot) {
    int e = threadIdx.x >> 5;      // 8 waves -> 8 experts
    int lane = threadIdx.x & 31;
    int running = 0;
    for (int base = 0; base < T_; base += 32) {
        int t = base + lane;
        int i0 = eidx[t * K_ + 0];
        int i1 = eidx[t * K_ + 1];
        bool chosen = (i0 == e) || (i1 == e);
        unsigned long long bal = __ballot(chosen);
        int before = __popcll(bal & ((1ull << lane) - 1ull));
        if (chosen) {
            int pos = running + before + 1;          // 1-based position in expert
            int slot = -1;
            if (pos <= C_) { slot = e * C_ + pos - 1; assign[slot] = t; }
            int k = (i0 == e) ? 0 : 1;
            tok_slot[t * K_ + k] = slot;             // -1 == dropped
        }
        running += __popcll(bal);
    }
}

// ---------------------------------------------------------------------------
// Kernel 2: per-token normalized affinity scales written to their slots
// ---------------------------------------------------------------------------
__global__ void moe_scale_kernel(const float* __restrict__ aff,
                                 const int* __restrict__ eidx,
                                 const int* __restrict__ tok_slot,
                                 float* __restrict__ scale_slot) {
    int t = blockIdx.x * blockDim.x + threadIdx.x;
    if (t >= T_) return;
    int e0 = eidx[t * K_ + 0], e1 = eidx[t * K_ + 1];
    int s0 = tok_slot[t * K_ + 0], s1 = tok_slot[t * K_ + 1];
    float v0 = (s0 >= 0) ? aff[t * E_ + e0] : 0.0f;
    float v1 = (s1 >= 0) ? aff[t * E_ + e1] : 0.0f;
    float s = fmaxf(fabsf(v0) + fabsf(v1), 1e-12f);   // NORMALIZE_TOPK
    if (s0 >= 0) scale_slot[s0] = v0 / s;
    if (s1 >= 0) scale_slot[s1] = v1 / s;
}

// ---------------------------------------------------------------------------
// Kernel 3: gather tokens into permuted (E*C, H) activation buffer
// ---------------------------------------------------------------------------
__global__ void moe_gather_kernel(const float* __restrict__ hs,
                                  const int* __restrict__ assign,
                                  float* __restrict__ xperm) {
    int row = blockIdx.x;                 // 0..E*C-1
    int tok = assign[row];
    float4* dst = (float4*)(xperm + (size_t)row * H_);
    float4 v = make_float4(0.f, 0.f, 0.f, 0.f);
    if (tok >= 0) v = ((const float4*)(hs + (size_t)tok * H_))[threadIdx.x];
    dst[threadIdx.x] = v;
}

// ---------------------------------------------------------------------------
// Kernel 4: grouped GEMM1 + SiLU gate.  Per expert: (C x H) @ (H x 2I),
// h = silu(gate)*up stored to hbuf (E*C, I).  128x128 tile / WG, 8 waves,
// each wave: 32(M) x 64(N) x {gate,up} in 16x16 WMMA f32 fragments.
// A tile kept row-major in LDS: fragment = one aligned ds_load_b64.
// ---------------------------------------------------------------------------
__global__ __launch_bounds__(256)
void moe_gemm1_kernel(const float* __restrict__ xperm,
                      const float* __restrict__ gup,
                      float* __restrict__ hbuf) {
    __shared__ float As[128][ASTRIDE];
    __shared__ float Bg[32][BSTRIDE];
    __shared__ float Bu[32][BSTRIDE];

    const int n0 = blockIdx.x * 128;          // gate col base (up uses n0 + I_)
    const int m0 = blockIdx.y * 128;
    const int e  = blockIdx.z;
    const int tid = threadIdx.x;
    const int lane = tid & 31;
    const int wid = tid >> 5;
    const int wm = wid >> 1;                  // 0..3 -> 32-row band
    const int wn = wid & 1;                   // 0..1 -> 64-col band
    const int lmod = lane & 15;
    const int khalf = (lane >> 4) * 2;        // A/B fragment K split

    const size_t arow0 = (size_t)(e * C_ + m0);
    const float* aptr = xperm + arow0 * H_;
    const float* gptr = gup + (size_t)e * H_ * (2 * I_);

    // staging coordinates for this thread (idx = tid + i*256, i=0..3)
    const int ar = tid >> 3;                  // A row (0..31 for i=0)
    const int ak = (tid & 7) * 4;             // A k offset
    const int bk = tid >> 5;                  // B k row (0..7 for i=0)
    const int bc = (tid & 31) * 4;            // B col offset

    v8f acc[2][4][2];
    #pragma unroll
    for (int mt = 0; mt < 2; ++mt)
        #pragma unroll
        for (int nt = 0; nt < 4; ++nt)
            #pragma unroll
            for (int g = 0; g < 2; ++g)
                #pragma unroll
                for (int r = 0; r < 8; ++r) acc[mt][nt][g][r] = 0.0f;

    for (int kc = 0; kc < H_; kc += 32) {
        // Stage A (128x32, row-major) and B gate/up (32x128 each)
        #pragma unroll
        for (int i = 0; i < 4; ++i) {
            int r  = ar + i * 32;
            stage16(aptr + (size_t)r * H_ + kc + ak, &As[r][ak]);
            int kk = bk + i * 8;
            const float* rowp = gptr + (size_t)(kc + kk) * (2 * I_);
            stage16(rowp + n0 + bc,      &Bg[kk][bc]);
            stage16(rowp + I_ + n0 + bc, &Bu[kk][bc]);
        }
#if !MOE_ASYNC_LDS
        if (kc + 32 < H_)   // global_prefetch path (sync staging only)
            __builtin_prefetch(gptr + (size_t)(kc + 32 + bk) * (2 * I_) + n0 + bc, 0, 1);
#endif
        stage_fence();

        #pragma unroll
        for (int kk = 0; kk < 32; kk += 4) {
            v2f a[2], bg[4], bu[4];
            #pragma unroll
            for (int mt = 0; mt < 2; ++mt) {
                int r = wm * 32 + mt * 16 + lmod;
                a[mt] = *(const v2f*)&As[r][kk + khalf];
            }
            #pragma unroll
            for (int nt = 0; nt < 4; ++nt) {
                int c = wn * 64 + nt * 16 + lmod;
                bg[nt].x = Bg[kk + khalf + 0][c];
                bg[nt].y = Bg[kk + khalf + 1][c];
                bu[nt].x = Bu[kk + khalf + 0][c];
                bu[nt].y = Bu[kk + khalf + 1][c];
            }
            #pragma unroll
            for (int mt = 0; mt < 2; ++mt)
                #pragma unroll
                for (int nt = 0; nt < 4; ++nt) {
                    acc[mt][nt][0] = __builtin_amdgcn_wmma_f32_16x16x4_f32(
                        false, a[mt], false, bg[nt], (short)0, acc[mt][nt][0], false, false);
                    acc[mt][nt][1] = __builtin_amdgcn_wmma_f32_16x16x4_f32(
                        false, a[mt], false, bu[nt], (short)0, acc[mt][nt][1], false, false);
                }
        }
        __syncthreads();
    }

    // Epilogue: h = silu(gate) * up
    const int lhalf8 = (lane >> 4) * 8;
    #pragma unroll
    for (int mt = 0; mt < 2; ++mt)
        #pragma unroll
        for (int nt = 0; nt < 4; ++nt)
            #pragma unroll
            for (int r = 0; r < 8; ++r) {
                int rowL = wm * 32 + mt * 16 + lhalf8 + r;
                int colL = wn * 64 + nt * 16 + lmod;
                float g = acc[mt][nt][0][r];
                float u = acc[mt][nt][1][r];
                float hv = (g / (1.0f + __expf(-g))) * u;
                hbuf[(arow0 + rowL) * (size_t)I_ + n0 + colL] = hv;
            }
}

// ---------------------------------------------------------------------------
// Kernel 5: grouped GEMM2 + scaled scatter-add.  Per expert: (C x I) @ (I x H),
// each output element atomically added into out[token] * scale.
// ---------------------------------------------------------------------------
__global__ __launch_bounds__(256)
void moe_gemm2_kernel(const float* __restrict__ hbuf,
                      const float* __restrict__ down,
                      const int* __restrict__ assign,
                      const float* __restrict__ scale_slot,
                      float* __restrict__ out) {
    __shared__ float As[128][ASTRIDE];
    __shared__ float Bd[32][BSTRIDE];

    const int n0 = blockIdx.x * 128;          // col in H
    const int m0 = blockIdx.y * 128;
    const int e  = blockIdx.z;
    const int tid = threadIdx.x;
    const int lane = tid & 31;
    const int wid = tid >> 5;
    const int wm = wid >> 1;
    const int wn = wid & 1;
    const int lmod = lane & 15;
    const int khalf = (lane >> 4) * 2;

    const size_t arow0 = (size_t)(e * C_ + m0);
    const float* aptr = hbuf + arow0 * I_;
    const float* dptr = down + (size_t)e * I_ * H_;

    const int ar = tid >> 3;
    const int ak = (tid & 7) * 4;
    const int bk = tid >> 5;
    const int bc = (tid & 31) * 4;

    v8f acc[2][4];
    #pragma unroll
    for (int mt = 0; mt < 2; ++mt)
        #pragma unroll
        for (int nt = 0; nt < 4; ++nt)
            #pragma unroll
            for (int r = 0; r < 8; ++r) acc[mt][nt][r] = 0.0f;

    for (int kc = 0; kc < I_; kc += 32) {
        #pragma unroll
        for (int i = 0; i < 4; ++i) {
            int r  = ar + i * 32;
            stage16(aptr + (size_t)r * I_ + kc + ak, &As[r][ak]);
            int kk = bk + i * 8;
            stage16(dptr + (size_t)(kc + kk) * H_ + n0 + bc, &Bd[kk][bc]);
        }
#if !MOE_ASYNC_LDS
        if (kc + 32 < I_)
            __builtin_prefetch(dptr + (size_t)(kc + 32 + bk) * H_ + n0 + bc, 0, 1);
#endif
        stage_fence();

        #pragma unroll
        for (int kk = 0; kk < 32; kk += 4) {
            v2f a[2], b[4];
            #pragma unroll
            for (int mt = 0; mt < 2; ++mt) {
                int r = wm * 32 + mt * 16 + lmod;
                a[mt] = *(const v2f*)&As[r][kk + khalf];
            }
            #pragma unroll
            for (int nt = 0; nt < 4; ++nt) {
                int c = wn * 64 + nt * 16 + lmod;
                b[nt].x = Bd[kk + khalf + 0][c];
                b[nt].y = Bd[kk + khalf + 1][c];
            }
            #pragma unroll
            for (int mt = 0; mt < 2; ++mt)
                #pragma unroll
                for (int nt = 0; nt < 4; ++nt)
                    acc[mt][nt] = __builtin_amdgcn_wmma_f32_16x16x4_f32(
                        false, a[mt], false, b[nt], (short)0, acc[mt][nt], false, false);
        }
        __syncthreads();
    }

    // Epilogue: scatter-add y * scale back to tokens
    const int lhalf8 = (lane >> 4) * 8;
    #pragma unroll
    for (int mt = 0; mt < 2; ++mt)
        #pragma unroll
        for (int nt = 0; nt < 4; ++nt)
            #pragma unroll
            for (int r = 0; r < 8; ++r) {
                int rowL = wm * 32 + mt * 16 + lhalf8 + r;
                int colL = wn * 64 + nt * 16 + lmod;
                int slot = e * C_ + m0 + rowL;
                int tok = assign[slot];
                if (tok >= 0) {
                    float v = acc[mt][nt][r] * scale_slot[slot];
                    atomicAdd(out + (size_t)tok * H_ + n0 + colL, v);
                }
            }
}

// ---------------------------------------------------------------------------
// Launcher
// ---------------------------------------------------------------------------
extern "C" void kernel_launch(void* const* d_in, const int* in_sizes, int n_in,
                              void* d_out, int out_size, void* d_ws, size_t ws_size,
                              hipStream_t stream) {
    (void)in_sizes; (void)n_in; (void)out_size; (void)ws_size;
    const float* hidden = (const float*)d_in[0];
    const float* affin  = (const float*)d_in[1];
    const int*   eidx   = (const int*)d_in[2];
    const float* gup    = (const float*)d_in[3];
    const float* down   = (const float*)d_in[4];
    float* out = (float*)d_out;

    // Workspace layout (~336 MB): xperm (67MB) | hbuf (268MB) | small arrays
    char* ws = (char*)d_ws;
    size_t off = 0;
    float* xperm = (float*)(ws + off);      off += (size_t)E_ * C_ * H_ * sizeof(float);
    float* hbuf  = (float*)(ws + off);      off += (size_t)E_ * C_ * I_ * sizeof(float);
    int*   assign = (int*)(ws + off);       off += (size_t)E_ * C_ * sizeof(int);
    int*   tok_slot = (int*)(ws + off);     off += (size_t)T_ * K_ * sizeof(int);
    float* scale_slot = (float*)(ws + off); off += (size_t)E_ * C_ * sizeof(float);

    moe_init_kernel<<<4096, 256, 0, stream>>>(out, assign, scale_slot);
    moe_route_kernel<<<1, 256, 0, stream>>>(eidx, assign, tok_slot);
    moe_scale_kernel<<<T_ / 256, 256, 0, stream>>>(affin, eidx, tok_slot, scale_slot);
    moe_gather_kernel<<<E_ * C_, 256, 0, stream>>>(hidden, assign, xperm);

    dim3 g1(I_ / 128, C_ / 128, E_);   // (32, 16, 8)
    moe_gemm1_kernel<<<g1, 256, 0, stream>>>(xperm, gup, hbuf);

    dim3 g2(H_ / 128, C_ / 128, E_);   // (8, 16, 8)
    moe_gemm2_kernel<<<g2, 256, 0, stream>>>(hbuf, down, assign, scale_slot, out);
}